// CombinedLoss_81784767250655
// MI455X (gfx1250) — compile-verified
//
#include <hip/hip_runtime.h>

typedef float v2f __attribute__((ext_vector_type(2)));
typedef float v8f __attribute__((ext_vector_type(8)));

#define DX    160
#define DXY   (160 * 160)
#define NB    (160 * 160 * 160)   // elements per batch (channel dim == 1)
#define NG    (NB / 4)            // float4 groups per batch
#define DX4   (DX / 4)            // groups per x-row
#define DXY4  (DXY / 4)           // groups per z-plane
#define GRIDX 1024                // blocks per batch
#define NBLK  (GRIDX * 4)         // total blocks

// Sum a value across all 32 lanes of a wave using the matrix pipe:
// A(16x4): lanes0-15 -> rows M, vgpr0=K0 (=v), vgpr1=K1 (=0); lanes16-31 -> K2(=v),K3(=0)
// B(4x16) = ones  =>  D[M][*] = v[M] + v[M+16]
// Sum of the 8 D vgprs gives half-sums per 16-lane group; shfl_xor(16) completes it.
__device__ __forceinline__ float wave_sum32_wmma(float v) {
  v2f a = {v, 0.0f};
  v2f b = {1.0f, 1.0f};
  v8f c = {};
  c = __builtin_amdgcn_wmma_f32_16x16x4_f32(false, a, false, b, (short)0, c,
                                            false, false);
  float s = ((c[0] + c[1]) + (c[2] + c[3])) + ((c[4] + c[5]) + (c[6] + c[7]));
  s += __shfl_xor(s, 16, 32);
  return s;  // total in every lane
}

__global__ __launch_bounds__(256) void combined_loss_partials(
    const float* __restrict__ in, const float* __restrict__ tg,
    float* __restrict__ part) {
  const int b  = blockIdx.y;
  const int bf = b * GRIDX + blockIdx.x;
  const float* xp = in + (size_t)b * NB;
  const float* tp = tg + (size_t)b * NB;

  float s_inter = 0.f, s_p = 0.f, s_t = 0.f, s_ce = 0.f, s_fo = 0.f, s_bd = 0.f;
  const float NEG_LOG_EPS = 27.631021f;  // -ln(1e-12f); 1.0f-1e-12f == 1.0f in fp32

  const int stride = GRIDX * 256;
  for (int g = blockIdx.x * 256 + threadIdx.x; g < NG; g += stride) {
    const int i = g * 4;
    if (g + stride < NG) {  // prefetch next grid-stride chunk (global_prefetch_b8)
      __builtin_prefetch(xp + i + stride * 4, 0, 0);
      __builtin_prefetch(tp + i + stride * 4, 0, 0);
    }
    const float4 xv = *(const float4*)(xp + i);
    const float4 tv = *(const float4*)(tp + i);

    // group coordinates (amortized over 4 elements)
    const int iz = g / DXY4;
    const int r  = g - iz * DXY4;
    const int iy = r / DX4;
    const int gx = r - iy * DX4;  // 0..39; element x = gx*4 + j

    const float xa[4] = {xv.x, xv.y, xv.z, xv.w};
    const float ta[4] = {tv.x, tv.y, tv.z, tv.w};

    // targets boundary bt = t - erode(t) for the 4 elements
    float bt4[4] = {0.f, 0.f, 0.f, 0.f};
    if ((ta[0] + ta[1] + ta[2] + ta[3]) != 0.0f) {
      const float4 z4  = {0.f, 0.f, 0.f, 0.f};
      const float4 tym = (iy > 0)      ? *(const float4*)(tp + i - DX)  : z4;
      const float4 typ = (iy < DX - 1) ? *(const float4*)(tp + i + DX)  : z4;
      const float4 tzm = (iz > 0)      ? *(const float4*)(tp + i - DXY) : z4;
      const float4 tzp = (iz < DX - 1) ? *(const float4*)(tp + i + DXY) : z4;
      const float  tl  = (gx > 0)        ? tp[i - 1] : 0.f;
      const float  tr  = (gx < DX4 - 1)  ? tp[i + 4] : 0.f;
      const float ym[4] = {tym.x, tym.y, tym.z, tym.w};
      const float yp[4] = {typ.x, typ.y, typ.z, typ.w};
      const float zm[4] = {tzm.x, tzm.y, tzm.z, tzm.w};
      const float zp[4] = {tzp.x, tzp.y, tzp.z, tzp.w};
      const float lf[4] = {tl, ta[0], ta[1], ta[2]};
      const float rt[4] = {ta[1], ta[2], ta[3], tr};
#pragma unroll
      for (int j = 0; j < 4; ++j) {
        float et = fminf(fminf(ta[j], fminf(lf[j], rt[j])),
                         fminf(fminf(ym[j], yp[j]), fminf(zm[j], zp[j])));
        bt4[j] = ta[j] - et;  // 0 automatically when center is 0
      }
    }

    const bool yzface =
        (iy == 0) | (iy == DX - 1) | (iz == 0) | (iz == DX - 1);

#pragma unroll
    for (int j = 0; j < 4; ++j) {
      const float x  = xa[j];
      const float t  = ta[j];
      const float ax = fabsf(x);
      const float e  = __expf(-ax);                      // v_exp_f32 path
      const float s  = __logf(1.0f + e);                 // log1p safe: e >= ~2e-3
      const float iv = __builtin_amdgcn_rcpf(1.0f + e);  // v_rcp_f32
      const float p  = (x >= 0.0f) ? iv : e * iv;        // sigmoid(x), stable

      // dice partials
      s_inter += p * t;
      s_p     += p;
      s_t     += t;

      // BCE element == -(t*log_sig(x) + (1-t)*log_sig(-x))
      const float ce = fmaxf(x, 0.0f) - x * t + s;
      s_ce += ce;

      // focal: alpha=1, gamma=2
      const float pt = p * t + (1.0f - p) * (1.0f - t);
      const float om = 1.0f - pt;
      s_fo += om * om * ce;

      // boundary: (p != 0) everywhere -> erosion of its mask is 0 on faces, 1 inside
      const bool face =
          yzface | ((gx == 0) & (j == 0)) | ((gx == DX4 - 1) & (j == 3));
      float term;
      if (face) {
        const float bic = fminf(fmaxf(p, 1e-12f), 1.0f);  // bi = p on faces
        term = (bt4[j] != 0.0f) ? -__logf(bic) : -__logf(1.0f - bic);
      } else {
        // bi = p-1 < 0 -> clipped to 1e-12; 1-1e-12 rounds to 1.0f -> log = 0
        term = bt4[j] * NEG_LOG_EPS;  // bt in {0,1}
      }
      s_bd += term;
    }
  }

  // wave32 reduction on the matrix pipe
  float rs[6];
  rs[0] = wave_sum32_wmma(s_inter);
  rs[1] = wave_sum32_wmma(s_p);
  rs[2] = wave_sum32_wmma(s_t);
  rs[3] = wave_sum32_wmma(s_ce);
  rs[4] = wave_sum32_wmma(s_fo);
  rs[5] = wave_sum32_wmma(s_bd);

  __shared__ float wsum[8][6];
  const int wave = threadIdx.x >> 5;
  const int lane = threadIdx.x & 31;
  if (lane == 0)
    for (int k = 0; k < 6; ++k) wsum[wave][k] = rs[k];
  __syncthreads();
  if (threadIdx.x == 0) {
    for (int k = 0; k < 6; ++k) {
      float v = 0.f;
      for (int w = 0; w < 8; ++w) v += wsum[w][k];
      part[k * NBLK + bf] = v;  // deterministic per-block partials
    }
  }
}

__global__ __launch_bounds__(256) void combined_loss_finalize(
    const float* __restrict__ part, float* __restrict__ out) {
  __shared__ float red[256];
  __shared__ float res[15];
  const int tid = threadIdx.x;

  // 12 per-batch dice reductions: k in {inter,psum,tsum} x b in 0..3
  for (int k = 0; k < 3; ++k) {
    for (int b = 0; b < 4; ++b) {
      float s = 0.0f;
      for (int j = tid; j < GRIDX; j += 256) s += part[k * NBLK + b * GRIDX + j];
      red[tid] = s;
      __syncthreads();
      for (int off = 128; off > 0; off >>= 1) {
        if (tid < off) red[tid] += red[tid + off];
        __syncthreads();
      }
      if (tid == 0) res[k * 4 + b] = red[0];
      __syncthreads();
    }
  }
  // 3 global reductions: ce, focal, boundary
  for (int k = 3; k < 6; ++k) {
    float s = 0.0f;
    for (int j = tid; j < NBLK; j += 256) s += part[k * NBLK + j];
    red[tid] = s;
    __syncthreads();
    for (int off = 128; off > 0; off >>= 1) {
      if (tid < off) red[tid] += red[tid + off];
      __syncthreads();
    }
    if (tid == 0) res[12 + (k - 3)] = red[0];
    __syncthreads();
  }

  if (tid == 0) {
    double dl = 0.0;
    for (int b = 0; b < 4; ++b) {
      const double inter = res[b], ps = res[4 + b], ts = res[8 + b];
      const double dice = (2.0 * inter + 1e-5) / (ps + ts + 1e-5);
      dl += 1.0 - dice;
    }
    dl *= 0.25;
    const double invN = 1.0 / (4.0 * (double)NB);
    const double ce = res[12] * invN;
    const double fo = res[13] * invN;
    const double bd = res[14] * invN;
    out[0] = (float)(0.4 * dl + 0.3 * ce + 0.2 * fo + 0.1 * bd);
  }
}

extern "C" void kernel_launch(void* const* d_in, const int* in_sizes, int n_in,
                              void* d_out, int out_size, void* d_ws,
                              size_t ws_size, hipStream_t stream) {
  const float* in = (const float*)d_in[0];
  const float* tg = (const float*)d_in[1];
  float* part = (float*)d_ws;  // 6 * NBLK floats = 96 KB

  dim3 grid(GRIDX, 4);
  combined_loss_partials<<<grid, 256, 0, stream>>>(in, tg, part);
  combined_loss_finalize<<<1, 256, 0, stream>>>(part, (float*)d_out);
}